// SPDClassifier_63642825392363
// MI455X (gfx1250) — compile-verified
//
#include <hip/hip_runtime.h>
#include <math.h>

// SPDNet classifier for MI455X (gfx1250, wave32, WMMA).
// All spectral ops (eig-clamp / eig-ReLU / log) are matrix polynomial
// iterations (Newton-Schulz sign & sqrt) -> dense f32 GEMMs on
// V_WMMA_F32_16X16X4_F32.
// Stage 1: one 200x200 matrix per workgroup, fully LDS-resident (2x160KB in
//          the 320KB WGP LDS). Emits h = W1 * relu_eps(sym(X)) * W1^T (64x64)
//          to workspace (needs 2048*64*64*4 = 32 MiB of d_ws).
// Stage 2: 64x64 eig-ReLU -> W2 bimap -> 32x32 eig-ReLU -> matrix log
//          (12 NS square roots + 4-term log series) -> triu feats -> MLP.
//
// OOB handling (no exec-mask churn anywhere):
//  - loads: indices CLAMPED before the k-loop (OOB A-rows/B-cols only affect
//           D-rows/cols that are never stored)
//  - stores: branchless address select -- OOB lanes store to a per-lane LDS
//            trash slot (v_cndmask + unconditional ds_store)

#define BATCH 2048
#define NREG  200
#define HID   64
#define HALFD 32
#define NFEAT 528
#define PROJ_EPS 1e-6f
#define RELU_THR 1e-4f

typedef float v2f __attribute__((ext_vector_type(2)));
typedef float v8f __attribute__((ext_vector_type(8)));

// ---- wave-level 16x16 f32 WMMA tile:  C(16x16) = A[mBase:,:K] * B[:K,nBase:]
__device__ __forceinline__ v8f tile_mm(const float* __restrict__ A, int lda,
                                       const float* __restrict__ B, int ldb,
                                       int mBase, int nBase, int K,
                                       int mMax, int nMax) {
  const int lane = threadIdx.x & 31;
  const int l16  = lane & 15;
  const int hh   = lane >> 4;
  int m = mBase + l16; if (m >= mMax) m = mMax - 1;   // clamp (see header note)
  int n = nBase + l16; if (n >= nMax) n = nMax - 1;
  const float* ap = A + m * lda + 2 * hh;
  const float* bp = B + 2 * hh * ldb + n;
  v8f acc = {};
#pragma unroll 4
  for (int k = 0; k < K; k += 4) {
    v2f av, bv;
    av.x = ap[0];
    av.y = ap[1];
    bv.x = bp[0];
    bv.y = bp[ldb];
    ap += 4;
    bp += 4 * ldb;
    acc = __builtin_amdgcn_wmma_f32_16x16x4_f32(false, av, false, bv,
                                                (short)0, acc, false, false);
  }
  return acc;
}

// Same, but B given transposed:  B_logical[k,n] = Bt[n*ldbt + k]
__device__ __forceinline__ v8f tile_mm_bt(const float* __restrict__ A, int lda,
                                          const float* __restrict__ Bt, int ldbt,
                                          int mBase, int nBase, int K,
                                          int mMax, int nMax) {
  const int lane = threadIdx.x & 31;
  const int l16  = lane & 15;
  const int hh   = lane >> 4;
  int m = mBase + l16; if (m >= mMax) m = mMax - 1;
  int n = nBase + l16; if (n >= nMax) n = nMax - 1;
  const float* ap = A + m * lda + 2 * hh;
  const float* bp = Bt + n * ldbt + 2 * hh;
  v8f acc = {};
#pragma unroll 4
  for (int k = 0; k < K; k += 4) {
    v2f av, bv;
    av.x = ap[0];
    av.y = ap[1];
    bv.x = bp[0];
    bv.y = bp[1];
    ap += 4;
    bp += 4;
    acc = __builtin_amdgcn_wmma_f32_16x16x4_f32(false, av, false, bv,
                                                (short)0, acc, false, false);
  }
  return acc;
}

// 32x32 macro-tile (2x2 WMMA tiles): 4 WMMAs share 2 A- and 2 B-fragments.
struct acc2x2 { v8f c00, c01, c10, c11; };
__device__ __forceinline__ acc2x2 tile_mm32(const float* __restrict__ A, int lda,
                                            const float* __restrict__ B, int ldb,
                                            int mBase, int nBase, int K,
                                            int mMax, int nMax) {
  const int lane = threadIdx.x & 31;
  const int l16  = lane & 15;
  const int hh   = lane >> 4;
  int m0 = mBase + l16;       if (m0 >= mMax) m0 = mMax - 1;
  int m1 = mBase + 16 + l16;  if (m1 >= mMax) m1 = mMax - 1;
  int n0 = nBase + l16;       if (n0 >= nMax) n0 = nMax - 1;
  int n1 = nBase + 16 + l16;  if (n1 >= nMax) n1 = nMax - 1;
  const float* ap0 = A + m0 * lda + 2 * hh;
  const float* ap1 = A + m1 * lda + 2 * hh;
  const float* bp  = B + 2 * hh * ldb;
  acc2x2 r;
  r.c00 = (v8f){}; r.c01 = (v8f){}; r.c10 = (v8f){}; r.c11 = (v8f){};
#pragma unroll 2
  for (int k = 0; k < K; k += 4) {
    v2f a0, a1, b0, b1;
    a0.x = ap0[0]; a0.y = ap0[1];
    a1.x = ap1[0]; a1.y = ap1[1];
    b0.x = bp[n0]; b0.y = bp[ldb + n0];
    b1.x = bp[n1]; b1.y = bp[ldb + n1];
    ap0 += 4; ap1 += 4; bp += 4 * ldb;
    r.c00 = __builtin_amdgcn_wmma_f32_16x16x4_f32(false, a0, false, b0, (short)0, r.c00, false, false);
    r.c01 = __builtin_amdgcn_wmma_f32_16x16x4_f32(false, a0, false, b1, (short)0, r.c01, false, false);
    r.c10 = __builtin_amdgcn_wmma_f32_16x16x4_f32(false, a1, false, b0, (short)0, r.c10, false, false);
    r.c11 = __builtin_amdgcn_wmma_f32_16x16x4_f32(false, a1, false, b1, (short)0, r.c11, false, false);
  }
  return r;
}

// Exact-fit store (no OOB possible): zero control flow.
__device__ __forceinline__ void tile_store_full(float* D, int ldd,
                                                int mBase, int nBase, v8f acc) {
  const int lane = threadIdx.x & 31;
  const int l16  = lane & 15;
  const int hh   = lane >> 4;
  float* dp = D + (mBase + 8 * hh) * ldd + nBase + l16;
#pragma unroll
  for (int jj = 0; jj < 8; ++jj) dp[jj * ldd] = acc[jj];
}

// Guarded store: OOB lanes are redirected to a per-lane LDS trash slot
// (branchless: address cndmask + unconditional ds_store).
__device__ __forceinline__ void tile_store_g(float* D, int ldd,
                                             int mBase, int nBase, v8f acc,
                                             int mMax, int nMax, float* trash) {
  const int lane = threadIdx.x & 31;
  const int l16  = lane & 15;
  const int hh   = lane >> 4;
  const int n = nBase + l16;
  const bool nok = n < nMax;
  float* tr = trash + lane;
#pragma unroll
  for (int jj = 0; jj < 8; ++jj) {
    int m = mBase + jj + 8 * hh;
    float* addr = (nok && m < mMax) ? (D + m * ldd + n) : tr;
    *addr = acc[jj];
  }
}

// Deterministic block sum: wave32 shuffle reduce + serial sum of wave partials.
__device__ __forceinline__ float block_sum(float v, float* prt, float* slot,
                                           int tid, int nthr) {
#pragma unroll
  for (int off = 16; off > 0; off >>= 1) v += __shfl_down(v, off, 32);
  if ((tid & 31) == 0) prt[tid >> 5] = v;
  __syncthreads();
  if (tid == 0) {
    float s = 0.0f;
    const int nw = nthr >> 5;
    for (int i = 0; i < nw; ++i) s += prt[i];
    slot[0] = s;
  }
  __syncthreads();
  return slot[0];
}

// ===================== Stage 1: 200x200 eig-clamp + W1 bimap =====================
__global__ __launch_bounds__(1024)
void spd_stage1(const float* __restrict__ X, const float* __restrict__ w1,
                float* __restrict__ hout) {
  extern __shared__ float smem[];
  float* Z     = smem;              // 40000: sign iterate (later Q in cols 0..63)
  float* P     = smem + 40000;      // 40000: Z^2 temp, then rebuilt B, then R
  float* prt   = smem + 80000;      // 32 wave partials
  float* scal  = smem + 80032;      // scalars
  float* trash = smem + 80040;      // 32 trash slots

  const int tid  = threadIdx.x;
  const int wave = tid >> 5;
  const int lane = tid & 31, l16 = lane & 15, hh = lane >> 4;
  const int b = blockIdx.x;
  const float* Xb = X + (size_t)b * (NREG * NREG);

  // B = sym(X) - eps*I  -> Z ; Frobenius norm
  float part = 0.0f;
  for (int idx = tid; idx < NREG * NREG; idx += 1024) {
    int i = idx / NREG, j = idx - i * NREG;
    float v = 0.5f * (Xb[i * NREG + j] + Xb[j * NREG + i]);
    if (i == j) v -= PROJ_EPS;
    Z[idx] = v;
    part += v * v;
  }
  const float fro2 = block_sum(part, prt, scal, tid, 1024);
  const float inv = 1.0f / (sqrtf(fro2) + 1e-30f);
  for (int idx = tid; idx < NREG * NREG; idx += 1024) Z[idx] *= inv;
  __syncthreads();

  // Newton-Schulz sign iterations: Z <- 1.5 Z - 0.5 Z^3  (in-place, row-phased)
  for (int it = 0; it < 20; ++it) {
    // P = Z*Z : 7x7 macro-tiles of 32x32 (2x2 WMMA each)
    for (int t = wave; t < 49; t += 32) {
      int ti = t / 7, tj = t - ti * 7;
      acc2x2 r = tile_mm32(Z, NREG, Z, NREG, ti * 32, tj * 32, NREG, NREG, NREG);
      tile_store_g(P, NREG, ti * 32,      tj * 32,      r.c00, NREG, NREG, trash);
      tile_store_g(P, NREG, ti * 32,      tj * 32 + 16, r.c01, NREG, NREG, trash);
      tile_store_g(P, NREG, ti * 32 + 16, tj * 32,      r.c10, NREG, NREG, trash);
      tile_store_g(P, NREG, ti * 32 + 16, tj * 32 + 16, r.c11, NREG, NREG, trash);
    }
    __syncthreads();
    // Z <- 1.5 Z - 0.5 Z*P : row-phased in-place (2 block-rows / phase)
    for (int ph = 0; ph < 7; ++ph) {
      int r  = ph * 2 + (wave >= 13 ? 1 : 0);
      int tj = (wave >= 13) ? wave - 13 : wave;
      bool active = (wave < 26) && (r < 13);
      v8f acc = {};
      if (active) acc = tile_mm(Z, NREG, P, NREG, r * 16, tj * 16, NREG, NREG, NREG);
      __syncthreads();
      if (active) {
        int n = tj * 16 + l16;
        bool nok = n < NREG;
        float* tr = trash + lane;
#pragma unroll
        for (int jj = 0; jj < 8; ++jj) {
          int m = r * 16 + jj + 8 * hh;
          float* addr = (nok && m < NREG) ? (Z + m * NREG + n) : tr;
          *addr = 1.5f * (*addr) - 0.5f * acc[jj];
        }
      }
      __syncthreads();
    }
  }

  // Rebuild B into P (re-read X; inputs unmodified)
  for (int idx = tid; idx < NREG * NREG; idx += 1024) {
    int i = idx / NREG, j = idx - i * NREG;
    float v = 0.5f * (Xb[i * NREG + j] + Xb[j * NREG + i]);
    if (i == j) v -= PROJ_EPS;
    P[idx] = v;
  }
  __syncthreads();

  // Q = (I + S) * W1^T  -> Z columns 0..63  (row-phased overwrite of S)
  for (int ph = 0; ph < 2; ++ph) {
    int t = ph * 32 + wave;
    bool active = t < 52;                    // 13 block-rows x 4 block-cols
    int r = t >> 2, c = t & 3;
    v8f acc = {};
    if (active) acc = tile_mm_bt(Z, NREG, w1, NREG, r * 16, c * 16, NREG, NREG, HID);
    __syncthreads();
    if (active) {
      int n = c * 16 + l16;                  // n < 64 always
      float* tr = trash + lane;
#pragma unroll
      for (int jj = 0; jj < 8; ++jj) {
        int m = r * 16 + jj + 8 * hh;
        int mc = (m < NREG) ? m : (NREG - 1);         // keep w1 read in-bounds
        float* addr = (m < NREG) ? (Z + m * NREG + n) : tr;
        *addr = w1[n * NREG + mc] + acc[jj];
      }
    }
    __syncthreads();
  }

  // R = B * Q  -> P columns 0..63  (row-phased overwrite of B)
  for (int ph = 0; ph < 2; ++ph) {
    int t = ph * 32 + wave;
    bool active = t < 52;
    int r = t >> 2, c = t & 3;
    v8f acc = {};
    if (active) acc = tile_mm(P, NREG, Z, NREG, r * 16, c * 16, NREG, NREG, HID);
    __syncthreads();
    if (active) {
      int n = c * 16 + l16;
      float* tr = trash + lane;
#pragma unroll
      for (int jj = 0; jj < 8; ++jj) {
        int m = r * 16 + jj + 8 * hh;
        float* addr = (m < NREG) ? (P + m * NREG + n) : tr;
        *addr = acc[jj];
      }
    }
    __syncthreads();
  }

  // h = 0.5 * W1 * R + eps*I  (uses W1*W1^T = I for the eps shift) -> global
  float* hb = hout + (size_t)b * (HID * HID);
  if (wave < 16) {
    int ti = wave >> 2, tj = wave & 3;
    v8f acc = tile_mm(w1, NREG, P, NREG, ti * 16, tj * 16, NREG, HID, HID);
    int n = tj * 16 + l16;
#pragma unroll
    for (int jj = 0; jj < 8; ++jj) {
      int m = ti * 16 + jj + 8 * hh;
      hb[m * HID + n] = 0.5f * acc[jj] + (m == n ? PROJ_EPS : 0.0f);
    }
  }
}

// ===================== Stage 2: 64x64 -> 32x32 -> logm -> MLP =====================
__global__ __launch_bounds__(256)
void spd_stage2(const float* __restrict__ hin, const float* __restrict__ w2,
                const float* __restrict__ lw1, const float* __restrict__ lb1,
                const float* __restrict__ lw2, const float* __restrict__ lb2,
                float* __restrict__ out) {
  extern __shared__ float smem[];
  float* b0 = smem;            // 4096 (64x64)
  float* b1 = smem + 4096;
  float* b2 = smem + 8192;
  float* b3 = smem + 12288;
  float* s0 = smem + 16384;    // six 32x32 buffers
  float* s1b = s0 + 1024;
  float* s2b = s0 + 2048;
  float* s3b = s0 + 3072;
  float* s4b = s0 + 4096;
  float* s5b = s0 + 5120;
  float* feats = s0 + 6144;    // 528
  float* zvec  = feats + 528;  // 32
  float* prt   = zvec + 32;    // 8 wave partials
  float* scal  = prt + 8;      // scalars

  const int tid  = threadIdx.x;
  const int wave = tid >> 5;
  const int lane = tid & 31, l16 = lane & 15, hh = lane >> 4;
  const int b = blockIdx.x;
  const float* hb = hin + (size_t)b * 4096;

  // ---- Hb = sym(h) - thr*I -> b2 ; frob norm
  float part = 0.0f;
  for (int idx = tid; idx < 4096; idx += 256) {
    int i = idx >> 6, j = idx & 63;
    float v = 0.5f * (hb[i * 64 + j] + hb[j * 64 + i]);
    if (i == j) v -= RELU_THR;
    b2[idx] = v;
    part += v * v;
  }
  float fro2 = block_sum(part, prt, scal, tid, 256);
  float inv = 1.0f / (sqrtf(fro2) + 1e-30f);
  for (int idx = tid; idx < 4096; idx += 256) b0[idx] = b2[idx] * inv;
  __syncthreads();

  // ---- NS sign on 64x64 (ping-pong b0 <-> b3, temp b1)
  float* z = b0; float* zt = b3; float* p = b1;
  for (int it = 0; it < 18; ++it) {
    for (int t = wave; t < 16; t += 8) {
      int ti = t >> 2, tj = t & 3;
      v8f acc = tile_mm(z, 64, z, 64, ti * 16, tj * 16, 64, 64, 64);
      tile_store_full(p, 64, ti * 16, tj * 16, acc);
    }
    __syncthreads();
    for (int t = wave; t < 16; t += 8) {
      int ti = t >> 2, tj = t & 3;
      v8f acc = tile_mm(z, 64, p, 64, ti * 16, tj * 16, 64, 64, 64);
      int n = tj * 16 + l16;
#pragma unroll
      for (int jj = 0; jj < 8; ++jj) {
        int m = ti * 16 + jj + 8 * hh;
        zt[m * 64 + n] = 1.5f * z[m * 64 + n] - 0.5f * acc[jj];
      }
    }
    __syncthreads();
    float* tmp = z; z = zt; zt = tmp;
  }
  // F = Hb*S -> p ; h64 = 0.5(Hb+F) + thr*I -> zt
  for (int t = wave; t < 16; t += 8) {
    int ti = t >> 2, tj = t & 3;
    v8f acc = tile_mm(b2, 64, z, 64, ti * 16, tj * 16, 64, 64, 64);
    tile_store_full(p, 64, ti * 16, tj * 16, acc);
  }
  __syncthreads();
  float* h64 = zt;
  for (int idx = tid; idx < 4096; idx += 256) {
    int i = idx >> 6, j = idx & 63;
    h64[idx] = 0.5f * (b2[idx] + p[idx]) + (i == j ? RELU_THR : 0.0f);
  }
  __syncthreads();

  // ---- V = h64 * W2^T (64x32) -> z-buffer, then g = W2 * V -> s0
  float* V = z;
  {
    int ti = wave >> 1, tj = wave & 1;  // 8 tiles, 8 waves
    v8f acc = tile_mm_bt(h64, 64, w2, 64, ti * 16, tj * 16, 64, 64, HALFD);
    tile_store_full(V, HALFD, ti * 16, tj * 16, acc);
  }
  __syncthreads();
  if (wave < 4) {
    int ti = wave >> 1, tj = wave & 1;
    v8f acc = tile_mm(w2, 64, V, HALFD, ti * 16, tj * 16, 64, HALFD, HALFD);
    tile_store_full(s0, HALFD, ti * 16, tj * 16, acc);
  }
  __syncthreads();

  // ---- 32x32 eig-ReLU:  Bk = sym(g) - thr*I -> s1b
  part = 0.0f;
  for (int idx = tid; idx < 1024; idx += 256) {
    int i = idx >> 5, j = idx & 31;
    float v = 0.5f * (s0[i * 32 + j] + s0[j * 32 + i]);
    if (i == j) v -= RELU_THR;
    s1b[idx] = v;
    part += v * v;
  }
  fro2 = block_sum(part, prt, scal, tid, 256);
  inv = 1.0f / (sqrtf(fro2) + 1e-30f);
  for (int idx = tid; idx < 1024; idx += 256) s2b[idx] = s1b[idx] * inv;
  __syncthreads();
  float* z2 = s2b; float* t2 = s4b; float* p2 = s3b;
  for (int it = 0; it < 18; ++it) {
    if (wave < 4) {
      int ti = wave >> 1, tj = wave & 1;
      v8f acc = tile_mm(z2, 32, z2, 32, ti * 16, tj * 16, 32, 32, 32);
      tile_store_full(p2, 32, ti * 16, tj * 16, acc);
    }
    __syncthreads();
    if (wave < 4) {
      int ti = wave >> 1, tj = wave & 1;
      v8f acc = tile_mm(z2, 32, p2, 32, ti * 16, tj * 16, 32, 32, 32);
      int n = tj * 16 + l16;
#pragma unroll
      for (int jj = 0; jj < 8; ++jj) {
        int m = ti * 16 + jj + 8 * hh;
        t2[m * 32 + n] = 1.5f * z2[m * 32 + n] - 0.5f * acc[jj];
      }
    }
    __syncthreads();
    float* tmp = z2; z2 = t2; t2 = tmp;
  }
  if (wave < 4) {                       // F2 = Bk * S2 -> p2
    int ti = wave >> 1, tj = wave & 1;
    v8f acc = tile_mm(s1b, 32, z2, 32, ti * 16, tj * 16, 32, 32, 32);
    tile_store_full(p2, 32, ti * 16, tj * 16, acc);
  }
  __syncthreads();
  // ghat = 0.5(Bk + F2) + thr*I -> s5b ; frob
  part = 0.0f;
  for (int idx = tid; idx < 1024; idx += 256) {
    int i = idx >> 5, j = idx & 31;
    float v = 0.5f * (s1b[idx] + p2[idx]) + (i == j ? RELU_THR : 0.0f);
    s5b[idx] = v;
    part += v * v;
  }
  fro2 = block_sum(part, prt, scal, tid, 256);

  // ---- matrix log: A2 = ghat / r ; 12 NS square roots ; 4-term series * 4096
  const float r = sqrtf(fro2) + 1e-30f;
  const float logScale = logf(r);
  inv = 1.0f / r;
  for (int idx = tid; idx < 1024; idx += 256) s0[idx] = s5b[idx] * inv;
  __syncthreads();

  float* Y = s0; float* Zi = s1b; float* Mm = s2b; float* Ta = s3b; float* Tb = s4b;
  for (int sq = 0; sq < 12; ++sq) {
    for (int idx = tid; idx < 1024; idx += 256) {
      int i = idx >> 5, j = idx & 31;
      Zi[idx] = (i == j) ? 1.0f : 0.0f;
    }
    __syncthreads();
    for (int it = 0; it < 14; ++it) {
      if (wave < 4) {                   // Mm = 1.5 I - 0.5 * Zi * Y
        int ti = wave >> 1, tj = wave & 1;
        v8f acc = tile_mm(Zi, 32, Y, 32, ti * 16, tj * 16, 32, 32, 32);
        int n = tj * 16 + l16;
#pragma unroll
        for (int jj = 0; jj < 8; ++jj) {
          int m = ti * 16 + jj + 8 * hh;
          Mm[m * 32 + n] = (m == n ? 1.5f : 0.0f) - 0.5f * acc[jj];
        }
      }
      __syncthreads();
      if (wave < 4) {                   // Ta = Y * Mm
        int ti = wave >> 1, tj = wave & 1;
        v8f acc = tile_mm(Y, 32, Mm, 32, ti * 16, tj * 16, 32, 32, 32);
        tile_store_full(Ta, 32, ti * 16, tj * 16, acc);
      } else {                          // Tb = Mm * Zi (waves 4..7 in parallel)
        int w = wave - 4;
        int ti = w >> 1, tj = w & 1;
        v8f acc = tile_mm(Mm, 32, Zi, 32, ti * 16, tj * 16, 32, 32, 32);
        tile_store_full(Tb, 32, ti * 16, tj * 16, acc);
      }
      __syncthreads();
      float* oY = Y; float* oZ = Zi; float* oM = Mm;
      Y = Ta; Zi = Tb; Mm = oY; Ta = oZ; Tb = oM;
    }
  }

  // E = Y - I ; L = E - E^2/2 + E^3/3 - E^4/4 ; LG = 4096*L + logScale*I
  float* E  = b0; float* E2 = b0 + 1024; float* E3 = b0 + 2048; float* E4 = b0 + 3072;
  float* LG = b1;
  for (int idx = tid; idx < 1024; idx += 256) {
    int i = idx >> 5, j = idx & 31;
    E[idx] = Y[idx] - (i == j ? 1.0f : 0.0f);
  }
  __syncthreads();
  if (wave < 4) {
    int ti = wave >> 1, tj = wave & 1;
    v8f acc = tile_mm(E, 32, E, 32, ti * 16, tj * 16, 32, 32, 32);
    tile_store_full(E2, 32, ti * 16, tj * 16, acc);
  }
  __syncthreads();
  if (wave < 4) {
    int ti = wave >> 1, tj = wave & 1;
    v8f acc = tile_mm(E, 32, E2, 32, ti * 16, tj * 16, 32, 32, 32);
    tile_store_full(E3, 32, ti * 16, tj * 16, acc);
  } else {
    int w = wave - 4;
    int ti = w >> 1, tj = w & 1;
    v8f acc = tile_mm(E2, 32, E2, 32, ti * 16, tj * 16, 32, 32, 32);
    tile_store_full(E4, 32, ti * 16, tj * 16, acc);
  }
  __syncthreads();
  for (int idx = tid; idx < 1024; idx += 256) {
    int i = idx >> 5, j = idx & 31;
    float l = E[idx] - 0.5f * E2[idx] + (1.0f / 3.0f) * E3[idx] - 0.25f * E4[idx];
    LG[idx] = 4096.0f * l + (i == j ? logScale : 0.0f);
  }
  __syncthreads();

  // ---- triu features + MLP
  for (int t = tid; t < 1024; t += 256) {
    int i = t >> 5, j = t & 31;
    if (j >= i) {
      int fidx = i * 32 - (i * (i - 1)) / 2 + (j - i);
      feats[fidx] = LG[t];
    }
  }
  __syncthreads();
  if (tid < 32) {
    float a = lb1[tid];
    for (int f = 0; f < NFEAT; ++f) a += feats[f] * lw1[tid * NFEAT + f];
    zvec[tid] = fmaxf(a, 0.0f);
  }
  __syncthreads();
  if (tid < 2) {
    float o = lb2[tid];
    for (int k = 0; k < 32; ++k) o += zvec[k] * lw2[tid * 32 + k];
    out[(size_t)b * 2 + tid] = o;
  }
}

extern "C" void kernel_launch(void* const* d_in, const int* in_sizes, int n_in,
                              void* d_out, int out_size, void* d_ws, size_t ws_size,
                              hipStream_t stream) {
  const float* X   = (const float*)d_in[0];
  const float* w1  = (const float*)d_in[1];
  const float* w2  = (const float*)d_in[2];
  const float* lw1 = (const float*)d_in[3];
  const float* lb1 = (const float*)d_in[4];
  const float* lw2 = (const float*)d_in[5];
  const float* lb2 = (const float*)d_in[6];
  float* out = (float*)d_out;
  float* hws = (float*)d_ws;  // needs 2048*64*64*4 = 32 MiB

  const size_t lds1 = (size_t)(2 * 40000 + 32 + 8 + 32) * sizeof(float);  // ~313 KB
  const size_t lds2 = (size_t)(16384 + 6 * 1024 + 528 + 32 + 8 + 8) * sizeof(float);

  spd_stage1<<<BATCH, 1024, lds1, stream>>>(X, w1, hws);
  spd_stage2<<<BATCH, 256, lds2, stream>>>(hws, w2, lw1, lb1, lw2, lb2, out);
}